// TypedMessagePassingLayer_80848464379988
// MI455X (gfx1250) — compile-verified
//
#include <hip/hip_runtime.h>

#define H 128
#define LN_EPS 1e-5f

typedef __attribute__((ext_vector_type(2))) float v2f;
typedef __attribute__((ext_vector_type(8))) float v8f;

// ---------------- kernel 0: zero the aggregation buffer ----------------
__global__ void zero_kernel(float* __restrict__ p, int n4) {
    int i = blockIdx.x * blockDim.x + threadIdx.x;
    if (i < n4) ((float4*)p)[i] = make_float4(0.f, 0.f, 0.f, 0.f);
}

// ---------------- kernel 1: gather + scatter-add ----------------
// one thread per (edge, 4-float chunk): 32 chunks per edge (H=128)
__global__ void scatter_kernel(const float* __restrict__ x,
                               const int* __restrict__ edge_index,
                               const int* __restrict__ edge_types,
                               const float* __restrict__ edge_emb,
                               float* __restrict__ agg, int E_) {
    int tid = blockIdx.x * blockDim.x + threadIdx.x;
    int e = tid >> 5;
    int c = tid & 31;
    if (e >= E_) return;
    int src = edge_index[e];
    int dst = edge_index[E_ + e];
    int t   = edge_types[e];
    float4 xv = ((const float4*)(x + (size_t)src * H))[c];
    float4 ev = ((const float4*)(edge_emb + (size_t)t * H))[c];
    float* d = agg + (size_t)dst * H + c * 4;
    atomicAdd(d + 0, xv.x + ev.x);
    atomicAdd(d + 1, xv.y + ev.y);
    atomicAdd(d + 2, xv.z + ev.z);
    atomicAdd(d + 3, xv.w + ev.w);
}

// ---------------- kernel 2: fused dual-GEMM + bias + relu + layernorm ----------------
// block = 256 threads = 8 wave32; each wave computes a 16x128 output tile.
// Weights (fp32) staged into LDS in WMMA B-fragment order, k4-major:
//   float index = ((k4*16 + mat*8 + jt)*32 + lane)*2 + e
// -> all 16 fragments of one k-step sit in one 4 KB chunk, so every inner
//    ds_load_b64 is base + small immediate (no per-frag VALU address math),
//    and lane*8B addressing is bank-conflict-free.
__global__ __launch_bounds__(256)
void fused_gemm_ln_kernel(const float* __restrict__ x,
                          const float* __restrict__ agg,
                          const float* __restrict__ W_self,
                          const float* __restrict__ b_self,
                          const float* __restrict__ W_msg,
                          const float* __restrict__ b_msg,
                          const float* __restrict__ gamma,
                          const float* __restrict__ beta,
                          float* __restrict__ out, int N_) {
    extern __shared__ float smem[];            // 32 k4-chunks * 1024 floats = 128 KB

    const int tid = threadIdx.x;

    // stage weights into k4-major fragment order (coalesced global reads)
    for (int idx = tid; idx < H * H; idx += 256) {
        int row = idx >> 7, col = idx & 127;    // W[row][col]: out_j = row, k = col
        int jt = row >> 4, n = row & 15;
        int k4 = col >> 2, rem = col & 3;
        int lane = ((rem >> 1) << 4) | n;       // half = rem>>1
        int e = rem & 1;
        int o = ((k4 * 16 + jt) * 32 + lane) * 2 + e;   // mat 0 slot
        smem[o]       = W_self[idx];
        smem[o + 512] = W_msg[idx];             // mat 1 slot: +8 frags = 512 floats
    }
    __syncthreads();

    const int lane = tid & 31;
    const int wave = tid >> 5;
    const int m    = lane & 15;                 // A-frag row / D-frag column-in-tile
    const int half = lane >> 4;
    const int base = blockIdx.x * 128 + wave * 16;

    // clamped source row for A fragments (keeps EXEC all-ones through WMMA)
    int rowA = base + m;
    if (rowA > N_ - 1) rowA = N_ - 1;
    const float* xr = x   + (size_t)rowA * H;
    const float* ar = agg + (size_t)rowA * H;

    v8f acc[8];
    #pragma unroll
    for (int jt = 0; jt < 8; ++jt) acc[jt] = {};

    // per-lane base into the fragment stream; advances 1024 floats per k-step
    const float* fb = smem + lane * 2;

    #pragma unroll 2
    for (int k4 = 0; k4 < 32; ++k4) {
        const int kk = k4 * 4 + half * 2;       // A layout: half0 -> K0,K1; half1 -> K2,K3
        v2f a0 = *(const v2f*)(xr + kk);        // x fragment
        v2f a1 = *(const v2f*)(ar + kk);        // aggregated fragment

        // batch all 16 B fragments (contiguous 4 KB chunk, immediate offsets)
        v2f b[16];
        #pragma unroll
        for (int f = 0; f < 16; ++f)
            b[f] = *(const v2f*)(fb + f * 64);

        #pragma unroll
        for (int jt = 0; jt < 8; ++jt) {
            acc[jt] = __builtin_amdgcn_wmma_f32_16x16x4_f32(
                false, a0, false, b[jt], (short)0, acc[jt], false, false);
            acc[jt] = __builtin_amdgcn_wmma_f32_16x16x4_f32(
                false, a1, false, b[8 + jt], (short)0, acc[jt], false, false);
        }
        fb += 1024;                              // next k4 chunk (4 KB)
    }

    // epilogue: bias + relu + layernorm
    float cb[8], cg[8], cB[8];
    #pragma unroll
    for (int jt = 0; jt < 8; ++jt) {
        int j = jt * 16 + m;                    // D column for this lane
        cb[jt] = b_self[j] + b_msg[j];
        cg[jt] = gamma[j];
        cB[jt] = beta[j];
    }

    #pragma unroll
    for (int v = 0; v < 8; ++v) {
        // D layout: VGPR v holds row (v + 8*half); row lives in one 16-lane half
        float vals[8];
        float s1 = 0.f, s2 = 0.f;
        #pragma unroll
        for (int jt = 0; jt < 8; ++jt) {
            float t = acc[jt][v] + cb[jt];
            t = fmaxf(t, 0.f);
            vals[jt] = t;
            s1 += t;
            s2 += t * t;
        }
        // reduce across the 16 lanes of this half (xor < 16 keeps bit4 fixed)
        #pragma unroll
        for (int off = 1; off < 16; off <<= 1) {
            s1 += __shfl_xor(s1, off, 32);
            s2 += __shfl_xor(s2, off, 32);
        }
        float mu  = s1 * (1.f / H);
        float var = s2 * (1.f / H) - mu * mu;
        float rs  = rsqrtf(var + LN_EPS);

        int row = base + v + half * 8;
        if (row < N_) {
            float* orow = out + (size_t)row * H;
            #pragma unroll
            for (int jt = 0; jt < 8; ++jt)
                orow[jt * 16 + m] = (vals[jt] - mu) * rs * cg[jt] + cB[jt];
        }
    }
}

extern "C" void kernel_launch(void* const* d_in, const int* in_sizes, int n_in,
                              void* d_out, int out_size, void* d_ws, size_t ws_size,
                              hipStream_t stream) {
    const float* x          = (const float*)d_in[0];
    const int*   edge_index = (const int*)  d_in[1];
    const int*   edge_types = (const int*)  d_in[2];
    const float* edge_emb   = (const float*)d_in[3];
    const float* W_self     = (const float*)d_in[4];
    const float* b_self     = (const float*)d_in[5];
    const float* W_msg      = (const float*)d_in[6];
    const float* b_msg      = (const float*)d_in[7];
    const float* gamma      = (const float*)d_in[8];
    const float* beta       = (const float*)d_in[9];
    float* out = (float*)d_out;
    float* agg = (float*)d_ws;                  // N*H fp32 aggregation buffer

    const int N_ = in_sizes[0] / H;
    const int E_ = in_sizes[1] / 2;

    // 1) zero aggregation buffer (d_ws is poisoned / must not carry state)
    int n4 = (N_ * H) / 4;
    zero_kernel<<<(n4 + 255) / 256, 256, 0, stream>>>(agg, n4);

    // 2) gather + scatter-add
    long long sthreads = (long long)E_ * 32;
    int sblocks = (int)((sthreads + 255) / 256);
    scatter_kernel<<<sblocks, 256, 0, stream>>>(x, edge_index, edge_types,
                                                edge_emb, agg, E_);

    // 3) fused dual-GEMM + epilogue, 128 KB dynamic LDS for weight fragments
    size_t lds_bytes = (size_t)32 * 1024 * sizeof(float);   // 131072 B
    hipFuncSetAttribute((const void*)fused_gemm_ln_kernel,
                        hipFuncAttributeMaxDynamicSharedMemorySize,
                        (int)lds_bytes);
    int gblocks = (N_ + 127) / 128;
    fused_gemm_ln_kernel<<<gblocks, 256, lds_bytes, stream>>>(
        x, agg, W_self, b_self, W_msg, b_msg, gamma, beta, out, N_);
}